// DistanceTransform_2293512536202
// MI455X (gfx1250) — compile-verified
//
#include <hip/hip_runtime.h>
#include <stdint.h>

#define W_DIM 2048

#define AS1 __attribute__((address_space(1)))
#define AS3 __attribute__((address_space(3)))

// Pointee type the builtin expects: 4 x i32 vector (per the clang diagnostic).
typedef int v4i __attribute__((vector_size(16)));

// ---------------------------------------------------------------------------
// Async global->LDS copy of 16 bytes (CDNA5 ASYNCcnt path).
// ---------------------------------------------------------------------------
__device__ __forceinline__ void async_copy_b128(const float* gsrc, float* ldst) {
#if __has_builtin(__builtin_amdgcn_global_load_async_to_lds_b128)
  __builtin_amdgcn_global_load_async_to_lds_b128(
      (AS1 v4i*)gsrc, (AS3 v4i*)ldst, 0, 0);
#else
  // Fallback: raw encoding. ldst points into the single __shared__ array of the
  // kernel, which is allocated at LDS offset 0, so its offset is recoverable
  // from the AS3 pointer value.
  unsigned loff = (unsigned)(size_t)(AS3 float*)ldst;
  unsigned long long ga = (unsigned long long)(size_t)gsrc;
  asm volatile("global_load_async_to_lds_b128 %0, %1, off"
               :: "v"(loff), "v"(ga) : "memory");
#endif
}

__device__ __forceinline__ void wait_async_lds() {
  asm volatile("s_wait_asynccnt 0" ::: "memory");
}

// ---------------------------------------------------------------------------
// K0: zero the global-max accumulator (ws is poisoned 0xAA by the harness and
// never re-poisoned, so we must reset it on every call).
// ---------------------------------------------------------------------------
__global__ void reset_ws_kernel(unsigned* ws) {
  if (threadIdx.x == 0 && blockIdx.x == 0) ws[0] = 0u;
}

// ---------------------------------------------------------------------------
// K1: per-column 1D pass (axis 0). One thread per column.
//   down pass: store (i - last0)      (last0 = index of last background above)
//   up pass:   d1 = min(stored, next0 - i);  f = mask ? d1*d1 : 0
// All arithmetic mirrors the fp32 reference (NEG=-1e9, POS=+1e9).
// ---------------------------------------------------------------------------
__global__ void edt_cols_kernel(const float* __restrict__ img,
                                float* __restrict__ f, int H, int W) {
  int j = blockIdx.x * blockDim.x + threadIdx.x;
  if (j >= W) return;

  float last = -1e9f;
  for (int i = 0; i < H; ++i) {
    if (i + 8 < H) __builtin_prefetch(&img[(size_t)(i + 8) * W + j], 0, 0);
    float v = img[(size_t)i * W + j];
    float fi = (float)i;
    if (!(v > 0.5f)) last = fi;            // background pixel
    f[(size_t)i * W + j] = fi - last;      // idx - last0 (fp32, matches ref)
  }

  float next = 1e9f;
  for (int i = H - 1; i >= 0; --i) {
    float v = img[(size_t)i * W + j];
    bool fg = (v > 0.5f);
    float fi = (float)i;
    if (!fg) next = fi;                    // next0 (scanning upward)
    float d1 = fminf(f[(size_t)i * W + j], next - fi);
    f[(size_t)i * W + j] = fg ? d1 * d1 : 0.0f;
  }
}

// ---------------------------------------------------------------------------
// K2: per-row exact min-plus convolution with the quadratic, then sqrt and
// block-wide max folded into a global atomic max. One block (256 threads)
// per row; the f row (8KB) is staged into LDS with async b128 copies.
// Thread t owns output columns j = t + 256*k, k = 0..7.
// ---------------------------------------------------------------------------
__global__ void edt_rows_kernel(float* __restrict__ f, unsigned* __restrict__ wsmax) {
  __shared__ __align__(16) float sf[W_DIM];

  const int t = threadIdx.x;
  float* frow = f + (size_t)blockIdx.x * W_DIM;

  // Stage the row into LDS: 256 lanes x 2 x 16B = 8192B.
  async_copy_b128(frow + t * 4,        &sf[t * 4]);
  async_copy_b128(frow + t * 4 + 1024, &sf[t * 4 + 1024]);
  wait_async_lds();
  __syncthreads();

  float best[8];
#pragma unroll
  for (int k = 0; k < 8; ++k) best[k] = 3.4e38f;

  const float dt = (float)t;
#pragma unroll 4
  for (int jp = 0; jp < W_DIM; ++jp) {
    float fv = sf[jp];                 // uniform across lanes -> LDS broadcast
    float d0 = dt - (float)jp;
#pragma unroll
    for (int k = 0; k < 8; ++k) {
      float dd = d0 + (float)(256 * k);
      best[k] = fminf(best[k], fmaf(dd, dd, fv));   // exact: sums < 2^24
    }
  }

  // sqrt + write back in place (LDS already holds the row; global row is ours)
  float bmax = 0.0f;
#pragma unroll
  for (int k = 0; k < 8; ++k) {
    float d = sqrtf(best[k]);
    frow[t + 256 * k] = d;
    bmax = fmaxf(bmax, d);
  }

  // Block max reduction (reuse sf after a barrier), then global bitwise
  // atomicMax — valid because all values are non-negative floats.
  __syncthreads();
  sf[t] = bmax;
  __syncthreads();
  for (int s = 128; s > 0; s >>= 1) {
    if (t < s) sf[t] = fmaxf(sf[t], sf[t + s]);
    __syncthreads();
  }
  if (t == 0) atomicMax(wsmax, __float_as_uint(sf[0]));
}

// ---------------------------------------------------------------------------
// K3: normalize by global max, truncate to uint8 (stored as float, the
// harness's representation of a uint8 output).
// ---------------------------------------------------------------------------
__global__ void finalize_kernel(float* __restrict__ f,
                                const unsigned* __restrict__ wsmax, int n) {
  int i = blockIdx.x * blockDim.x + threadIdx.x;
  if (i >= n) return;
  float m = __uint_as_float(*wsmax);
  float d = f[i];
  if (m > 0.0f) d = d / m * 255.0f;      // same op order as reference
  f[i] = (float)(unsigned char)d;        // C truncation == numpy astype(uint8)
}

// ---------------------------------------------------------------------------
extern "C" void kernel_launch(void* const* d_in, const int* in_sizes, int n_in,
                              void* d_out, int out_size, void* d_ws, size_t ws_size,
                              hipStream_t stream) {
  const float* img = (const float*)d_in[0];
  float* buf = (float*)d_out;          // used as f-scratch, then final output
  unsigned* wsmax = (unsigned*)d_ws;   // 4-byte global max accumulator

  const int total = in_sizes[0];
  const int W = W_DIM;
  const int H = total / W;

  reset_ws_kernel<<<1, 32, 0, stream>>>(wsmax);
  edt_cols_kernel<<<(W + 255) / 256, 256, 0, stream>>>(img, buf, H, W);
  edt_rows_kernel<<<H, 256, 0, stream>>>(buf, wsmax);
  finalize_kernel<<<(total + 255) / 256, 256, 0, stream>>>(buf, wsmax, total);
}